// Simple2DRNN_83846351552890
// MI455X (gfx1250) — compile-verified
//
#include <hip/hip_runtime.h>
#include <cmath>

#define T_STEPS  4096
#define B_TOTAL  4096
#define WAVE     32
#define CHUNK    16                 // timesteps staged per LDS buffer
#define NBUF     3                  // pipeline depth: 3*16 = 48 <= ASYNCcnt max 63
#define B2       (B_TOTAL * 2)      // floats per timestep row (8192)
#define ROWBYTES (B2 * 4)           // 32768 bytes per timestep row

__device__ __forceinline__ float rnn_tanh(float v) {
#if __has_builtin(__builtin_amdgcn_tanhf)
    return __builtin_amdgcn_tanhf(v);          // v_tanh_f32 (gfx1250 TRANS)
#elif __has_builtin(__builtin_amdgcn_tanh_f32)
    return __builtin_amdgcn_tanh_f32(v);
#else
    return tanhf(v);
#endif
}

// One wave32 per block, one lane per batch element. Single-wave blocks keep the
// async triple-buffer pipeline barrier-free (only this wave touches its LDS).
__global__ __launch_bounds__(WAVE) void
rnn2d_async_kernel(const float* __restrict__ x, const float* __restrict__ h0,
                   const float* __restrict__ Wih, const float* __restrict__ bih,
                   const float* __restrict__ Whh,
                   const float* __restrict__ Wfc, const float* __restrict__ bfc,
                   float* __restrict__ out)
{
    __shared__ float xs[NBUF][CHUNK][WAVE * 2];   // 12 KB: three 16-step x tiles

    const int tid = threadIdx.x;
    const int b   = blockIdx.x * WAVE + tid;

    // Uniform 2x2 weights / biases -> scalar registers (lane-uniform loads).
    const float w00 = Wih[0], w01 = Wih[1], w10 = Wih[2], w11 = Wih[3];
    const float u00 = Whh[0], u01 = Whh[1], u10 = Whh[2], u11 = Whh[3];
    const float f00 = Wfc[0], f01 = Wfc[1], f10 = Wfc[2], f11 = Wfc[3];
    const float c0  = bih[0], c1  = bih[1];
    const float d0  = bfc[0], d1  = bfc[1];

    float h0v = h0[b * 2 + 0];
    float h1v = h0[b * 2 + 1];

    // Raw LDS byte address of this lane's column (generic LDS addr truncates to
    // the workgroup-relative LDS offset in bits [31:0], per ISA aperture rules).
    const unsigned ldsbase  = (unsigned)(size_t)(&xs[0][0][0]) + (unsigned)(tid * 8);
    const unsigned bufbytes = CHUNK * WAVE * 8;
    const unsigned goff0    = (unsigned)(b * 8);   // lane's byte offset at t=0

    const int nchunks = T_STEPS / CHUNK;           // 256

    // ---- prologue: async-stage chunks 0..2 (16 x b64 per lane each) ----
#pragma unroll
    for (int k = 0; k < NBUF; ++k) {
        unsigned l = ldsbase + (unsigned)k * bufbytes;
        unsigned g = goff0 + (unsigned)k * (unsigned)(CHUNK * ROWBYTES);
#pragma unroll
        for (int tt = 0; tt < CHUNK; ++tt) {
            asm volatile("global_load_async_to_lds_b64 %0, %1, %2"
                         :: "v"(l), "v"(g), "s"(x) : "memory");
            l += WAVE * 8;
            g += ROWBYTES;
        }
    }

    int buf = 0;                                   // chunk c lives in buf = c % 3
    for (int c = 0; c < nchunks; ++c) {
        // Async loads complete in order; allow the (up to) two younger chunks
        // to stay outstanding while guaranteeing chunk c has landed in LDS.
        if (c <= nchunks - 3) {
            asm volatile("s_wait_asynccnt %0" :: "i"(2 * CHUNK) : "memory");
        } else if (c == nchunks - 2) {
            asm volatile("s_wait_asynccnt %0" :: "i"(CHUNK) : "memory");
        } else {
            asm volatile("s_wait_asynccnt 0x0" ::: "memory");
        }

#pragma unroll
        for (int tt = 0; tt < CHUNK; ++tt) {
            const float2 xv = *(const float2*)&xs[buf][tt][tid * 2]; // ds_load_b64
            // x contribution (off the h critical path)
            const float a0 = fmaf(xv.x, w00, fmaf(xv.y, w01, c0));
            const float a1 = fmaf(xv.x, w10, fmaf(xv.y, w11, c1));
            // recurrence critical path: 2 dependent FMAs + v_tanh_f32
            const float p0 = fmaf(h0v, u00, fmaf(h1v, u01, a0));
            const float p1 = fmaf(h0v, u10, fmaf(h1v, u11, a1));
            h0v = rnn_tanh(p0);
            h1v = rnn_tanh(p1);
            const float y0 = fmaf(h0v, f00, fmaf(h1v, f01, d0));
            const float y1 = fmaf(h0v, f10, fmaf(h1v, f11, d1));
            // NT store: out is write-once, keep it out of L2 so x stays resident.
            union { double d; float f[2]; } pk;
            pk.f[0] = y0; pk.f[1] = y1;
            const long t = (long)c * CHUNK + tt;
            __builtin_nontemporal_store(pk.d, (double*)(out + t * (long)B2 + b * 2));
        }

        // Refill the buffer we just consumed with chunk c+3.
        if (c + NBUF < nchunks) {
            // WAR guard: ds_loads from this buffer must retire before the async
            // engine overwrites it.
            asm volatile("s_wait_dscnt 0x0" ::: "memory");
            unsigned l = ldsbase + (unsigned)buf * bufbytes;
            unsigned g = goff0 + (unsigned)(c + NBUF) * (unsigned)(CHUNK * ROWBYTES);
#pragma unroll
            for (int tt = 0; tt < CHUNK; ++tt) {
                asm volatile("global_load_async_to_lds_b64 %0, %1, %2"
                             :: "v"(l), "v"(g), "s"(x) : "memory");
                l += WAVE * 8;
                g += ROWBYTES;
            }
        }
        buf = (buf == NBUF - 1) ? 0 : buf + 1;
    }

    // final hidden state appended after out: (1, B, 2)
    float* hf = out + (long)T_STEPS * B2;
    *(float2*)(hf + b * 2) = make_float2(h0v, h1v);
}

extern "C" void kernel_launch(void* const* d_in, const int* in_sizes, int n_in,
                              void* d_out, int out_size, void* d_ws, size_t ws_size,
                              hipStream_t stream) {
    (void)in_sizes; (void)n_in; (void)d_ws; (void)ws_size; (void)out_size;
    const float* x   = (const float*)d_in[0];
    const float* h0  = (const float*)d_in[1];
    const float* Wih = (const float*)d_in[2];
    const float* bih = (const float*)d_in[3];
    const float* Whh = (const float*)d_in[4];
    const float* Wfc = (const float*)d_in[5];
    const float* bfc = (const float*)d_in[6];
    float* out = (float*)d_out;

    dim3 grid(B_TOTAL / WAVE);   // 128 single-wave workgroups, spread across WGPs
    dim3 block(WAVE);
    hipLaunchKernelGGL(rnn2d_async_kernel, grid, block, 0, stream,
                       x, h0, Wih, bih, Whh, Wfc, bfc, out);
}